// MultiHeadAttention_52991306498287
// MI455X (gfx1250) — compile-verified
//
#include <hip/hip_runtime.h>
#include <hip/hip_bf16.h>

#define B_ 16
#define S_ 1024
#define D_ 16
#define H_ 4
#define DEPTH_ 4

typedef float v2f __attribute__((ext_vector_type(2)));
typedef float v8f __attribute__((ext_vector_type(8)));

__device__ __forceinline__ float wave_max32(float x) {
#pragma unroll
  for (int o = 16; o > 0; o >>= 1) x = fmaxf(x, __shfl_xor(x, o, 32));
  return x;
}
__device__ __forceinline__ float wave_sum32(float x) {
#pragma unroll
  for (int o = 16; o > 0; o >>= 1) x += __shfl_xor(x, o, 32);
  return x;
}

// ---------------------------------------------------------------------------
// Input projection: OUT[b,h,s,d] = (X @ W + bias) split to heads.
// X: [B*S, 16] row-major.  One wave handles 16 rows via 4x WMMA f32 16x16x4.
// ---------------------------------------------------------------------------
__global__ void __launch_bounds__(32) proj_in_kernel(const float* __restrict__ X,
                                                     const float* __restrict__ W,
                                                     const float* __restrict__ bias,
                                                     float* __restrict__ outp) {
  const int row0 = blockIdx.x * 16;
  const int lane = threadIdx.x & 31;
  const int half = lane >> 4;
  const int n    = lane & 15;   // serves as A-row m and B-col n
  const int off  = half * 2;    // K element pair selector

  v8f acc = {};
#pragma unroll
  for (int c4 = 0; c4 < 4; ++c4) {
    const int k0 = c4 * 4;
    float2 av = *(const float2*)(X + (size_t)(row0 + n) * 16 + k0 + off);
    v2f a; a[0] = av.x; a[1] = av.y;
    v2f bf; bf[0] = W[(k0 + off) * 16 + n]; bf[1] = W[(k0 + off + 1) * 16 + n];
    acc = __builtin_amdgcn_wmma_f32_16x16x4_f32(false, a, false, bf, (short)0, acc,
                                                false, false);
  }
  const float bn = bias[n];
#pragma unroll
  for (int i = 0; i < 8; ++i) {
    const int m  = i + half * 8;
    const int sg = row0 + m;          // global row in [0, B*S)
    const int bI = sg >> 10;
    const int s  = sg & 1023;
    // split heads: column n -> head n>>2, depth n&3 ; layout [B,H,S,4]
    outp[(((size_t)bI * H_ + (n >> 2)) * S_ + s) * DEPTH_ + (n & 3)] = acc[i] + bn;
  }
}

// ---------------------------------------------------------------------------
// Output projection, IN PLACE on merged heads: M[r,:] = M[r,:] @ W + bias.
// Safe in place: one wave per 16-row tile, all loads precede all stores in
// wave program order, tiles are disjoint across blocks.
// ---------------------------------------------------------------------------
__global__ void __launch_bounds__(32) proj_out_kernel(float* __restrict__ M,
                                                      const float* __restrict__ W,
                                                      const float* __restrict__ bias) {
  const int row0 = blockIdx.x * 16;
  const int lane = threadIdx.x & 31;
  const int half = lane >> 4;
  const int n    = lane & 15;
  const int off  = half * 2;

  v8f acc = {};
#pragma unroll
  for (int c4 = 0; c4 < 4; ++c4) {
    const int k0 = c4 * 4;
    float2 av = *(const float2*)(M + (size_t)(row0 + n) * 16 + k0 + off);
    v2f a; a[0] = av.x; a[1] = av.y;
    v2f bf; bf[0] = W[(k0 + off) * 16 + n]; bf[1] = W[(k0 + off + 1) * 16 + n];
    acc = __builtin_amdgcn_wmma_f32_16x16x4_f32(false, a, false, bf, (short)0, acc,
                                                false, false);
  }
  const float bn = bias[n];
#pragma unroll
  for (int i = 0; i < 8; ++i) {
    const int m = i + half * 8;
    M[(size_t)(row0 + m) * 16 + n] = acc[i] + bn;
  }
}

// ---------------------------------------------------------------------------
// Fused attention. grid = B * (S/16) blocks of 256 threads (8 wave32).
// Each block: load mask tiles (byte flags) once, then for 3 variants x 4 heads:
//   WMMA logits (16 x 1024) -> LDS, masked; rowwise softmax; NT-store w;
//   accumulate out = w @ v; store merged att into d_out out-slot.
// ---------------------------------------------------------------------------
__global__ void __launch_bounds__(256) attn_kernel(
    const float* __restrict__ mask,
    const float* __restrict__ mqp, const float* __restrict__ mkp,
    const float* __restrict__ mvp, const float* __restrict__ dqp,
    const float* __restrict__ dkp, const float* __restrict__ dvp,
    float* __restrict__ dout) {
  __shared__ float         lg[16 * S_];        // 64 KB logits
  __shared__ unsigned char mqs[16 * S_];       // 16 KB pre-softmax mask (q rows)
  __shared__ unsigned char mts[S_ * 17];       // 17 KB transposed post mask, pad 17

  const int b    = blockIdx.x >> 6;
  const int q0   = (blockIdx.x & 63) * 16;
  const int tid  = threadIdx.x;
  const int lane = tid & 31;
  const int wave = tid >> 5;
  const int half = lane >> 4;
  const int n    = lane & 15;
  const int off  = half * 2;

  // ---- mask tiles -> LDS (once per block, reused 12x) ----
  const float* mb = mask + (size_t)b * S_ * S_ + (size_t)q0 * S_;  // rows q0..q0+15
  for (int idx = tid; idx < 4096; idx += 256) {
    float4 g = ((const float4*)mb)[idx];
    const int base = idx * 4;
    mqs[base + 0] = (g.x != 0.f);
    mqs[base + 1] = (g.y != 0.f);
    mqs[base + 2] = (g.z != 0.f);
    mqs[base + 3] = (g.w != 0.f);
  }
  const float* mtb = mask + (size_t)b * S_ * S_ + q0;  // element (k, j): mtb[k*S + j]
  for (int idx = tid; idx < 4096; idx += 256) {
    const int k = idx >> 2, part = (idx & 3) * 4;
    float4 g = *(const float4*)(mtb + (size_t)k * S_ + part);
    const int base = k * 17 + part;
    mts[base + 0] = (g.x != 0.f);
    mts[base + 1] = (g.y != 0.f);
    mts[base + 2] = (g.z != 0.f);
    mts[base + 3] = (g.w != 0.f);
  }
  __syncthreads();

  const float* Qs[3] = {mqp, mqp, dqp};
  const float* Ks[3] = {mkp, dkp, dkp};
  const float* Vs[3] = {mvp, dvp, dvp};
  const size_t wOffs[3] = {262144ull, 67633152ull, 135004160ull};
  const size_t oOffs[3] = {0ull, 67371008ull, 134742016ull};

  for (int v = 0; v < 3; ++v) {
    for (int h = 0; h < H_; ++h) {
      const size_t bh = (size_t)(b * H_ + h) * S_;
      const float* qb = Qs[v] + (bh + q0) * DEPTH_;
      const float* kb = Ks[v] + bh * DEPTH_;
      const float* vb = Vs[v] + bh * DEPTH_;

      // A-frag: Q tile 16x4 (row n, K pair off/off+1)
      float2 aq = *(const float2*)(qb + n * DEPTH_ + off);
      v2f a; a[0] = aq.x; a[1] = aq.y;

      // logits: 64 k-tiles of 16, 8 per wave, one WMMA each (K=4 == depth)
      for (int kt = wave; kt < 64; kt += 8) {
        float2 bkv = *(const float2*)(kb + (size_t)(kt * 16 + n) * DEPTH_ + off);
        v2f bf; bf[0] = bkv.x; bf[1] = bkv.y;
        v8f c = {};
        c = __builtin_amdgcn_wmma_f32_16x16x4_f32(false, a, false, bf, (short)0, c,
                                                  false, false);
        const int kcol = kt * 16 + n;
#pragma unroll
        for (int i = 0; i < 8; ++i) {
          const int q = i + half * 8;
          lg[q * S_ + kcol] = c[i] * 0.5f + (mqs[q * S_ + kcol] ? -1e9f : 0.f);
        }
      }
      __syncthreads();

      // rowwise softmax + post-mask + NT store of w + out accumulation
      for (int rr = wave; rr < 16; rr += 8) {
        const float* lrow = lg + rr * S_;
        float mx = -3.4e38f;
#pragma unroll 4
        for (int j = 0; j < 32; ++j) mx = fmaxf(mx, lrow[lane + 32 * j]);
        mx = wave_max32(mx);
        float sm = 0.f;
#pragma unroll 4
        for (int j = 0; j < 32; ++j) sm += __expf(lrow[lane + 32 * j] - mx);
        sm = wave_sum32(sm);
        const float inv = 1.f / sm;

        float a0 = 0.f, a1 = 0.f, a2 = 0.f, a3 = 0.f;
        float* wrow = dout + wOffs[v] + (bh + q0 + rr) * S_;
#pragma unroll 4
        for (int j = 0; j < 32; ++j) {
          const int k = lane + 32 * j;
          float e = __expf(lrow[k] - mx) * inv;
          e = mts[k * 17 + rr] ? 0.f : e;            // w *= (1 - mask^T)
          __builtin_nontemporal_store(e, wrow + k);  // 805MB stream, bypass reuse
          float4 vv = *(const float4*)(vb + (size_t)k * DEPTH_);
          a0 += e * vv.x; a1 += e * vv.y; a2 += e * vv.z; a3 += e * vv.w;
        }
        a0 = wave_sum32(a0); a1 = wave_sum32(a1);
        a2 = wave_sum32(a2); a3 = wave_sum32(a3);
        if (lane == 0) {
          float4 o; o.x = a0; o.y = a1; o.z = a2; o.w = a3;
          // merged-heads layout directly into the out_* slot (projected later)
          *(float4*)(dout + oOffs[v] + (size_t)(b * S_ + q0 + rr) * D_ + h * DEPTH_) = o;
        }
      }
      __syncthreads();
    }
  }
}

extern "C" void kernel_launch(void* const* d_in, const int* in_sizes, int n_in,
                              void* d_out, int out_size, void* d_ws, size_t ws_size,
                              hipStream_t stream) {
  const float* med  = (const float*)d_in[0];
  const float* diag = (const float*)d_in[1];
  const float* mask = (const float*)d_in[2];
  const float* wmq = (const float*)d_in[3];  const float* bmq = (const float*)d_in[4];
  const float* wmk = (const float*)d_in[5];  const float* bmk = (const float*)d_in[6];
  const float* wmv = (const float*)d_in[7];  const float* bmv = (const float*)d_in[8];
  const float* wdq = (const float*)d_in[9];  const float* bdq = (const float*)d_in[10];
  const float* wdk = (const float*)d_in[11]; const float* bdk = (const float*)d_in[12];
  const float* wdv = (const float*)d_in[13]; const float* bdv = (const float*)d_in[14];
  const float* wo  = (const float*)d_in[15]; const float* bo  = (const float*)d_in[16];

  float* ws = (float*)d_ws;
  const size_t PN = (size_t)B_ * H_ * S_ * DEPTH_;  // 262144 floats per projection
  float* mqp = ws + 0 * PN;
  float* mkp = ws + 1 * PN;
  float* mvp = ws + 2 * PN;
  float* dqp = ws + 3 * PN;
  float* dkp = ws + 4 * PN;
  float* dvp = ws + 5 * PN;
  float* out = (float*)d_out;

  const int TILES = (B_ * S_) / 16;  // 1024

  proj_in_kernel<<<TILES, 32, 0, stream>>>(med,  wmq, bmq, mqp);
  proj_in_kernel<<<TILES, 32, 0, stream>>>(med,  wmk, bmk, mkp);
  proj_in_kernel<<<TILES, 32, 0, stream>>>(med,  wmv, bmv, mvp);
  proj_in_kernel<<<TILES, 32, 0, stream>>>(diag, wdq, bdq, dqp);
  proj_in_kernel<<<TILES, 32, 0, stream>>>(diag, wdk, bdk, dkp);
  proj_in_kernel<<<TILES, 32, 0, stream>>>(diag, wdv, bdv, dvp);

  attn_kernel<<<B_ * (S_ / 16), 256, 0, stream>>>(mask, mqp, mkp, mvp, dqp, dkp, dvp, out);

  proj_out_kernel<<<TILES, 32, 0, stream>>>(out + 0ull,         wo, bo);  // out_mm
  proj_out_kernel<<<TILES, 32, 0, stream>>>(out + 67371008ull,  wo, bo);  // out_md
  proj_out_kernel<<<TILES, 32, 0, stream>>>(out + 134742016ull, wo, bo);  // out_dd
}